// QuantumInspiredRL_12850542149978
// MI455X (gfx1250) — compile-verified
//
#include <hip/hip_runtime.h>
#include <hip/hip_bf16.h>

// ---------------------------------------------------------------------------
// QuantumInspiredRL for MI455X (gfx1250, wave32)
//
// Roofline: GEMM state[8192x8192] @ enc_w^T[8192x64] = 8.6 GFLOP over a
// 256 MB stream -> ~11 us at 23.3 TB/s. Must run on V_WMMA_F32_16X16X4_F32.
// All post-GEMM work reduces to operations on a single 64-vector because the
// gates are linear and the measurement sums over batch AND qubit dims.
// ---------------------------------------------------------------------------

typedef __attribute__((ext_vector_type(2))) float v2f;
typedef __attribute__((ext_vector_type(4))) float f4;
typedef __attribute__((ext_vector_type(8))) float v8f;

#define BATCH 8192
#define SDIM  8192
#define QDIM  64
#define ADIM  1024
#define HDIM  64
#define NGATE (3 * QDIM * 3)   // L * Q * 3 = 576

// ---- kernel 1: WMMA GEMM -> partial logits -------------------------------
#define KP    128              // K piece staged in LDS
#define WPB   4                // waves (16-row strips) per block
#define BSTR  (KP + 4)         // LDS row stride (dwords): 16B aligned, no bank conflicts

__global__ __launch_bounds__(WPB * 32)
void k_gemm(const float* __restrict__ state,
            const float* __restrict__ encw,
            float* __restrict__ ppart,   // [KS][BATCH][QDIM]
            int kchunk)                  // SDIM / KS
{
    __shared__ float sB[QDIM * BSTR];

    const int tid  = threadIdx.x;
    const int lane = tid & 31;
    const int wv   = tid >> 5;
    const int lo   = lane & 15;     // M (for A) / N (for B) within tile
    const int hi   = lane >> 4;     // K half selector
    const int row0 = (blockIdx.x * WPB + wv) * 16;
    const int kc0  = blockIdx.y * kchunk;

    v8f acc0 = {}, acc1 = {}, acc2 = {}, acc3 = {};

    // A layout (16x4 f32): lanes 0-15 hold K={0,1} in v[0],v[1]; lanes 16-31 K={2,3}
    const float* aptr = state + (size_t)(row0 + lo) * SDIM + kc0 + hi * 2;

    for (int kp0 = 0; kp0 < kchunk; kp0 += KP) {
        __syncthreads();
        // Cooperative stage of enc_w piece [QDIM][KP] into LDS (coalesced b128).
        for (int i = tid; i < QDIM * (KP / 4); i += WPB * 32) {
            int q  = i >> 5;            // KP/4 == 32 float4 per row
            int kg = i & 31;
            f4 v = *(const f4*)(encw + (size_t)q * SDIM + kc0 + kp0 + kg * 4);
            *(f4*)(&sB[q * BSTR + kg * 4]) = v;
        }
        __syncthreads();

        if (kp0 + KP < kchunk)
            __builtin_prefetch(aptr + kp0 + KP, 0, 1);   // global_prefetch for A stream

        #pragma unroll 8
        for (int kk = 0; kk < KP; kk += 4) {
            v2f av = *(const v2f*)(aptr + kp0 + kk);
            // B layout (4x16 f32): lane n = lo, lanes 0-15 K={0,1}, lanes 16-31 K={2,3}
            v2f bv0 = *(const v2f*)(&sB[(0 * 16 + lo) * BSTR + kk + hi * 2]);
            v2f bv1 = *(const v2f*)(&sB[(1 * 16 + lo) * BSTR + kk + hi * 2]);
            v2f bv2 = *(const v2f*)(&sB[(2 * 16 + lo) * BSTR + kk + hi * 2]);
            v2f bv3 = *(const v2f*)(&sB[(3 * 16 + lo) * BSTR + kk + hi * 2]);
            acc0 = __builtin_amdgcn_wmma_f32_16x16x4_f32(false, av, false, bv0, (short)0, acc0, false, false);
            acc1 = __builtin_amdgcn_wmma_f32_16x16x4_f32(false, av, false, bv1, (short)0, acc1, false, false);
            acc2 = __builtin_amdgcn_wmma_f32_16x16x4_f32(false, av, false, bv2, (short)0, acc2, false, false);
            acc3 = __builtin_amdgcn_wmma_f32_16x16x4_f32(false, av, false, bv3, (short)0, acc3, false, false);
        }
    }

    // C/D layout: VGPR j -> M = j + 8*hi, N = lo (lanes 0-15) / lo (16-31)
    float* dst = ppart + (size_t)blockIdx.y * BATCH * QDIM;
    #pragma unroll
    for (int j = 0; j < 8; ++j) {
        size_t m = (size_t)(row0 + j + 8 * hi);
        dst[m * QDIM + 0 * 16 + lo] = acc0[j];
        dst[m * QDIM + 1 * 16 + lo] = acc1[j];
        dst[m * QDIM + 2 * 16 + lo] = acc2[j];
        dst[m * QDIM + 3 * 16 + lo] = acc3[j];
    }
}

// ---- kernel 2: softmax per row + deterministic column-sum -----------------
__global__ __launch_bounds__(256)
void k_softmax(const float* __restrict__ ppart,
               const float* __restrict__ encb,
               float* __restrict__ colsum,   // [gridDim.x][QDIM]
               int ks)
{
    __shared__ float sw[8 * QDIM];
    const int tid  = threadIdx.x;
    const int lane = tid & 31;
    const int wv   = tid >> 5;
    const size_t b = (size_t)blockIdx.x * 256 + tid;

    float e[QDIM];
    #pragma unroll
    for (int q = 0; q < QDIM; ++q) e[q] = encb[q];
    for (int c = 0; c < ks; ++c) {
        const float* p = ppart + ((size_t)c * BATCH + b) * QDIM;
        #pragma unroll
        for (int q = 0; q < QDIM; ++q) e[q] += p[q];
    }
    float mx = -3.0e38f;
    #pragma unroll
    for (int q = 0; q < QDIM; ++q) mx = fmaxf(mx, e[q]);
    float sum = 0.0f;
    #pragma unroll
    for (int q = 0; q < QDIM; ++q) { e[q] = __expf(e[q] - mx); sum += e[q]; }
    float inv = 1.0f / sum;

    // butterfly reduce each component across the wave (deterministic order)
    #pragma unroll
    for (int q = 0; q < QDIM; ++q) {
        float v = e[q] * inv;
        #pragma unroll
        for (int m = 16; m >= 1; m >>= 1) v += __shfl_xor(v, m, 32);
        e[q] = v;
    }
    if (lane == 0) {
        #pragma unroll
        for (int q = 0; q < QDIM; ++q) sw[wv * QDIM + q] = e[q];
    }
    __syncthreads();
    if (tid < QDIM) {
        float s = 0.0f;
        #pragma unroll
        for (int w = 0; w < 8; ++w) s += sw[w * QDIM + tid];
        colsum[(size_t)blockIdx.x * QDIM + tid] = s;
    }
}

// ---- kernel 3: gates on the 64-vector + measurement + MLP -----------------
__global__ __launch_bounds__(1024)
void k_epilogue(const float* __restrict__ colsum, int nblk,
                const float* __restrict__ qparams,
                const float* __restrict__ mops,   // [A][Q][2]
                const float* __restrict__ w1, const float* __restrict__ bb1,
                const float* __restrict__ w2, const float* __restrict__ bb2,
                float* __restrict__ out)
{
    __shared__ float sv[QDIM];
    __shared__ float sh[HDIM];
    __shared__ float sm[ADIM];
    const int tid  = threadIdx.x;
    const int lane = tid & 31;

    if (tid < QDIM) {
        float s = 0.0f;
        for (int k = 0; k < nblk; ++k) s += colsum[(size_t)k * QDIM + tid];
        sv[tid] = s;
    }
    __syncthreads();

    // 576 sequential gates on the 64-vector; v[i] = c*v[i] + s*v[i-1] (cyclic).
    // Vector split over wave 0: a = v[lane], b = v[32+lane].
    if (tid < 32) {
        float a = sv[lane], b = sv[32 + lane];
        for (int t = 0; t < NGATE; ++t) {
            float th = qparams[t] * 0.5f;
            float c = __cosf(th), s = __sinf(th);
            float ra  = __shfl(a, (lane + 31) & 31, 32);
            float rb  = __shfl(b, (lane + 31) & 31, 32);
            float a31 = __shfl(a, 31, 32);
            float b31 = __shfl(b, 31, 32);
            float na = c * a + s * ((lane == 0) ? b31 : ra);
            float nb = c * b + s * ((lane == 0) ? a31 : rb);
            a = na; b = nb;
        }
        sv[lane] = a; sv[32 + lane] = b;
    }
    __syncthreads();

    {   // measurements[a] = sum_q mops[a][q][0] * v[q]
        float s = 0.0f;
        const float* mp = mops + (size_t)tid * QDIM * 2;
        #pragma unroll
        for (int q = 0; q < QDIM; ++q) s += mp[q * 2] * sv[q];
        sm[tid] = s;
    }
    __syncthreads();
    if (tid < HDIM) {
        float s = bb1[tid];
        const float* wp = w1 + (size_t)tid * ADIM;
        for (int a = 0; a < ADIM; ++a) s += wp[a] * sm[a];
        sh[tid] = fmaxf(s, 0.0f);
    }
    __syncthreads();
    {
        float s = bb2[tid];
        const float* wp = w2 + (size_t)tid * HDIM;
        #pragma unroll
        for (int j = 0; j < HDIM; ++j) s += wp[j] * sh[j];
        out[tid] = s;
    }
}

// ---- host launcher --------------------------------------------------------
extern "C" void kernel_launch(void* const* d_in, const int* in_sizes, int n_in,
                              void* d_out, int out_size, void* d_ws, size_t ws_size,
                              hipStream_t stream) {
    const float* state = (const float*)d_in[0];
    const float* encw  = (const float*)d_in[1];
    const float* encb  = (const float*)d_in[2];
    const float* qp    = (const float*)d_in[3];
    const float* mo    = (const float*)d_in[4];
    const float* w1    = (const float*)d_in[5];
    const float* b1    = (const float*)d_in[6];
    const float* w2    = (const float*)d_in[7];
    const float* b2    = (const float*)d_in[8];
    float* out = (float*)d_out;

    const int nblk2 = BATCH / 256;                       // 32 softmax blocks
    const size_t colsum_bytes = (size_t)nblk2 * QDIM * sizeof(float);
    const size_t per_chunk    = (size_t)BATCH * QDIM * sizeof(float);

    int ks = 1;                                          // K-split for occupancy
    if      (ws_size >= colsum_bytes + 8 * per_chunk) ks = 8;
    else if (ws_size >= colsum_bytes + 4 * per_chunk) ks = 4;
    else if (ws_size >= colsum_bytes + 2 * per_chunk) ks = 2;

    float* ppart  = (float*)d_ws;
    float* colsum = (float*)((char*)d_ws + (size_t)ks * per_chunk);
    const int kchunk = SDIM / ks;

    dim3 g1(BATCH / 16 / WPB, ks);                       // 128 x ks blocks
    k_gemm<<<g1, WPB * 32, 0, stream>>>(state, encw, ppart, kchunk);
    k_softmax<<<nblk2, 256, 0, stream>>>(ppart, encb, colsum, ks);
    k_epilogue<<<1, 1024, 0, stream>>>(colsum, nblk2, qp, mo, w1, b1, w2, b2, out);
}